// _MeshConv_62388694942533
// MI455X (gfx1250) — compile-verified
//
#include <hip/hip_runtime.h>

// MeshConv for MI455X (gfx1250, wave32).
// Pipeline:
//   k_transpose : x [B,CIN,NV] -> xT [NV, 256]  (channel-contiguous for gathers)
//   k_facegrad  : per-face EW/NS-projected gradients -> gegn [NF, 256, 2]
//   k_feat      : per-vertex feature assembly -> feat4 [NV, 256, 4] = {x, lap, gv_ew, gv_ns}
//                 (Laplacian 7-gather + F2V 6-gather fused, coalesced float4 stores)
//   k_gemm      : out[b,o,n] = W[64x128] @ feat[128 x n] via V_WMMA_F32_16X16X4_F32.
//                 Branchless B-fragment: one global_load_b64 per lane per K-step
//                 (lanes 0-15 -> K rows 0,1 = {x,lap}; lanes 16-31 -> K rows 2,3 =
//                 {gv_ew,gv_ns}), matching the 16x16x4 B VGPR layout. One wave owns
//                 all four 16-row o-tiles so each B fetch feeds 4 WMMAs.
// Workspace: NV*256 + NF*256*2 + NV*256*4 floats = 377,497,600 bytes.

#define NVERT 40962
#define NFACE 81920
#define NBATCH 8
#define CINCH 32
#define COUTCH 64
#define NCH 256                  // NBATCH * CINCH
#define XT_FLOATS   (NVERT * NCH)          // 10,486,272
#define GEGN_FLOATS (NFACE * NCH * 2)      // 41,943,040
#define NTILES 2561              // ceil(NVERT / 16)

typedef __attribute__((ext_vector_type(2))) float v2f;
typedef __attribute__((ext_vector_type(4))) float v4f;
typedef __attribute__((ext_vector_type(8))) float v8f;

__global__ void k_transpose(const float* __restrict__ x, float* __restrict__ xT) {
  __shared__ float tile[32][33];
  int v0 = blockIdx.x * 32;
  int c0 = blockIdx.y * 32;
  int tx = threadIdx.x, ty = threadIdx.y;
  int v = v0 + tx;
  if (v < NVERT) tile[ty][tx] = x[(c0 + ty) * NVERT + v];
  __syncthreads();
  int vw = v0 + ty;
  if (vw < NVERT) xT[vw * NCH + c0 + tx] = tile[tx][ty];
}

__global__ void k_facegrad(const float* __restrict__ xT,
                           const float* __restrict__ Gv,
                           const int* __restrict__ G_cols,
                           const float* __restrict__ NS,
                           const float* __restrict__ EW,
                           float* __restrict__ gegn) {
  int f = blockIdx.x;       // one face per block
  int bc = threadIdx.x;     // channel (b*32+i), 256 threads
  float ge = 0.f, gn = 0.f;
#pragma unroll
  for (int k = 0; k < 3; ++k) {
    int base = 3 * (k * NFACE + f);
    float acc = 0.f;
#pragma unroll
    for (int j = 0; j < 3; ++j)
      acc += Gv[base + j] * xT[G_cols[base + j] * NCH + bc];
    ge += EW[f * 3 + k] * acc;
    gn += NS[f * 3 + k] * acc;
  }
  v2f r; r.x = ge; r.y = gn;
  *(v2f*)(gegn + (f * NCH + bc) * 2) = r;
}

// Assemble feat4[v][bc][4] = { x, laplacian, gv_ew, gv_ns }.
__global__ void k_feat(const float* __restrict__ xT,
                       const float* __restrict__ gegn,
                       const float* __restrict__ Lv,
                       const int* __restrict__ L_cols,
                       const float* __restrict__ Fv,
                       const int* __restrict__ F_cols,
                       float* __restrict__ feat4) {
  int v = blockIdx.x;       // one vertex per block
  int bc = threadIdx.x;
  float lap = 0.f;
#pragma unroll
  for (int j = 0; j < 7; ++j)
    lap += Lv[v * 7 + j] * xT[L_cols[v * 7 + j] * NCH + bc];
  float se = 0.f, sn = 0.f;
#pragma unroll
  for (int j = 0; j < 6; ++j) {
    float w = Fv[v * 6 + j];
    v2f g = *(const v2f*)(gegn + (F_cols[v * 6 + j] * NCH + bc) * 2);
    se += w * g.x;
    sn += w * g.y;
  }
  v4f r;
  r.x = xT[v * NCH + bc];   // d=0: x
  r.y = lap;                // d=1: laplacian
  r.z = se;                 // d=2: gv_ew
  r.w = sn;                 // d=3: gv_ns
  *(v4f*)(feat4 + (v * NCH + bc) * 4) = r;
}

__global__ void __launch_bounds__(256)
k_gemm(const float* __restrict__ feat4,
       const float* __restrict__ coeffs,   // [64, 32, 4] == [64, 128]
       const float* __restrict__ bias,     // [64]
       float* __restrict__ out) {          // [B, 64, NV]
  __shared__ __align__(16) float W[COUTCH * 128];
  int tid = threadIdx.x;
  for (int k = tid; k < COUTCH * 128; k += 256) W[k] = coeffs[k];
  __syncthreads();

  int wave = blockIdx.x * 8 + (tid >> 5);   // 8 waves/block
  int lane = tid & 31;
  int b  = wave / NTILES;                   // batch
  int nt = wave - b * NTILES;               // 16-vertex tile
  int v0 = nt * 16;
  int ln = lane & 15;
  bool hi = lane >= 16;
  int vtx = v0 + ln; if (vtx >= NVERT) vtx = NVERT - 1;   // clamp ragged edge
  int koff = hi ? 2 : 0;
  // Per-lane feat stream: 512 contiguous bytes over the 32-step K loop.
  int frow = (vtx * NCH + b * CINCH) * 4 + koff;

  // C/D layout: VGPR j holds M=j (lanes 0-15) and M=j+8 (lanes 16-31), N=lane%16.
  v8f c0, c1, c2, c3;
#pragma unroll
  for (int j = 0; j < 8; ++j) {
    int m = j + (hi ? 8 : 0);
    c0[j] = bias[m];
    c1[j] = bias[16 + m];
    c2[j] = bias[32 + m];
    c3[j] = bias[48 + m];
  }

  // K = 128 = 32 channels x 4 feature types; one WMMA step == one channel i.
#pragma unroll 4
  for (int i = 0; i < CINCH; ++i) {
    // B fragment (4x16): lanes 0-15 -> K=0,1 ({x,lap}); 16-31 -> K=2,3 ({gv_ew,gv_ns}).
    v2f bf = *(const v2f*)(feat4 + frow + i * 4);
    // A fragments (16x4): lane L<16 -> A[L, K=0..1], L>=16 -> A[L-16, K=2..3].
    int kb = i * 4 + koff;
    v2f a0 = *(const v2f*)(&W[(ln)      * 128 + kb]);
    v2f a1 = *(const v2f*)(&W[(16 + ln) * 128 + kb]);
    v2f a2 = *(const v2f*)(&W[(32 + ln) * 128 + kb]);
    v2f a3 = *(const v2f*)(&W[(48 + ln) * 128 + kb]);
    c0 = __builtin_amdgcn_wmma_f32_16x16x4_f32(false, a0, false, bf, (short)0, c0, false, false);
    c1 = __builtin_amdgcn_wmma_f32_16x16x4_f32(false, a1, false, bf, (short)0, c1, false, false);
    c2 = __builtin_amdgcn_wmma_f32_16x16x4_f32(false, a2, false, bf, (short)0, c2, false, false);
    c3 = __builtin_amdgcn_wmma_f32_16x16x4_f32(false, a3, false, bf, (short)0, c3, false, false);
  }

  int n = v0 + ln;
  if (n < NVERT) {
    int mb = hi ? 8 : 0;
#pragma unroll
    for (int j = 0; j < 8; ++j) {
      out[(b * COUTCH + 0  + mb + j) * NVERT + n] = c0[j];
      out[(b * COUTCH + 16 + mb + j) * NVERT + n] = c1[j];
      out[(b * COUTCH + 32 + mb + j) * NVERT + n] = c2[j];
      out[(b * COUTCH + 48 + mb + j) * NVERT + n] = c3[j];
    }
  }
}

extern "C" void kernel_launch(void* const* d_in, const int* in_sizes, int n_in,
                              void* d_out, int out_size, void* d_ws, size_t ws_size,
                              hipStream_t stream) {
  const float* x      = (const float*)d_in[0];
  const float* coeffs = (const float*)d_in[1];
  const float* bias   = (const float*)d_in[2];
  const float* Gv     = (const float*)d_in[3];
  const float* Lv     = (const float*)d_in[4];
  const float* Fv     = (const float*)d_in[5];
  const float* NS     = (const float*)d_in[6];
  const float* EW     = (const float*)d_in[7];
  const int* G_cols   = (const int*)d_in[9];    // G_rows unused (fixed 3/row)
  const int* L_cols   = (const int*)d_in[11];   // L_rows unused (fixed 7/row)
  const int* F_cols   = (const int*)d_in[13];   // F_rows unused (fixed 6/row)
  float* out = (float*)d_out;

  float* xT    = (float*)d_ws;
  float* gegn  = xT + XT_FLOATS;
  float* feat4 = gegn + (size_t)GEGN_FLOATS;

  dim3 tgrid((NVERT + 31) / 32, NCH / 32);
  k_transpose<<<tgrid, dim3(32, 32), 0, stream>>>(x, xT);
  k_facegrad<<<NFACE, 256, 0, stream>>>(xT, Gv, G_cols, NS, EW, gegn);
  k_feat<<<NVERT, 256, 0, stream>>>(xT, gegn, Lv, L_cols, Fv, F_cols, feat4);
  k_gemm<<<NTILES, 256, 0, stream>>>(feat4, coeffs, bias, out);
}